// cross_entropy_with_hnm_for_one_class_detection_24352464568686
// MI455X (gfx1250) — compile-verified
//
#include <hip/hip_runtime.h>
#include <hip/hip_bf16.h>
#include <cstdint>

#define TILE 512

struct BranchStats {
  unsigned pos_num;      // 0
  unsigned neg_num;      // 1
  float    sum_pos_ce;   // 2
  float    sum_mask;     // 3
  float    sum_sq;       // 4
  unsigned rr;           // 5  remaining rank among positive candidates
  unsigned prefix;       // 6  radix-select prefix
  unsigned done;         // 7
  unsigned thr_bits;     // 8  float bits of threshold
  unsigned cnt_neg;      // 9
  float    sum_neg_ce;   // 10
  unsigned pad;          // 11
};
#define STATS_U32 ((int)(sizeof(BranchStats) / 4))

// ---------------- TDM (Tensor Data Mover) ----------------
#if defined(__has_builtin)
#if __has_builtin(__builtin_amdgcn_tensor_load_to_lds)
#define HAVE_TDM 1
#endif
#endif

typedef unsigned int tdm_u4 __attribute__((ext_vector_type(4)));
typedef int          tdm_i8 __attribute__((ext_vector_type(8)));
typedef int          tdm_i4 __attribute__((ext_vector_type(4)));

__device__ __forceinline__ void wait_tensorcnt0() {
#if defined(__has_builtin) && __has_builtin(__builtin_amdgcn_s_wait_tensorcnt)
  __builtin_amdgcn_s_wait_tensorcnt(0);
#else
  asm volatile("s_wait_tensorcnt 0x0" ::: "memory");
#endif
}

#ifdef HAVE_TDM
// Load a 2D tile (tile_w f32 elements wide, tile_h rows, row stride in elements)
// from global memory at gsrc into LDS byte address lds_addr.
__device__ __forceinline__ void tdm_load_2d_f32(unsigned lds_addr, const float* gsrc,
                                                unsigned tile_w, unsigned tile_h,
                                                unsigned long long row_stride) {
  unsigned long long ga = (unsigned long long)(uintptr_t)gsrc;
  tdm_u4 g0;
  g0.x = 1u;                                         // count=1, user descriptor
  g0.y = lds_addr;                                   // lds_addr (bytes)
  g0.z = (unsigned)(ga & 0xFFFFFFFFull);             // global_addr[31:0]
  g0.w = (unsigned)((ga >> 32) & 0x01FFFFFFull)      // global_addr[56:32]
       | (2u << 30);                                 // type = 2 ("image")
  unsigned td0 = (unsigned)row_stride;               // tensor_dim0 (elements)
  unsigned td1 = tile_h;                             // tensor_dim1
  tdm_i8 g1;
  g1[0] = (int)(2u << 16);                                           // data_size=4B
  g1[1] = (int)((td0 & 0xFFFFu) << 16);                              // tensor_dim0 lo16
  g1[2] = (int)(((td0 >> 16) & 0xFFFFu) | ((td1 & 0xFFFFu) << 16));  // dim0 hi | dim1 lo
  g1[3] = (int)(((td1 >> 16) & 0xFFFFu) | ((tile_w & 0xFFFFu) << 16)); // dim1 hi | tile_dim0
  g1[4] = (int)(tile_h & 0xFFFFu);                                   // tile_dim1 (tile_dim2=0)
  g1[5] = (int)(unsigned)(row_stride & 0xFFFFFFFFull);               // dim0_stride lo32
  g1[6] = (int)(unsigned)((row_stride >> 32) & 0xFFFFull);           // dim0_stride hi16
  g1[7] = 0;
  tdm_i4 gz = {0, 0, 0, 0};
#if __clang_major__ >= 23
  tdm_i8 gz8 = {0, 0, 0, 0, 0, 0, 0, 0};
  __builtin_amdgcn_tensor_load_to_lds(g0, g1, gz, gz, gz8, 0);
#else
  __builtin_amdgcn_tensor_load_to_lds(g0, g1, gz, gz, 0);
#endif
}
#endif

// ---------------- kernels ----------------

__global__ void init_kernel(unsigned* p, int n) {
  for (int i = blockIdx.x * blockDim.x + threadIdx.x; i < n; i += gridDim.x * blockDim.x)
    p[i] = 0u;
}

// Pass 1: softmax, flags, pos-CE, bbox sums, packed p_neg. TDM-staged into LDS.
__global__ __launch_bounds__(TILE) void pass1_kernel(
    const float* __restrict__ ps, const float* __restrict__ pb,
    const float* __restrict__ gm, const float* __restrict__ gl,
    unsigned* __restrict__ packed, BranchStats* __restrict__ st,
    int HW, int chunks) {
  __shared__ float tile[16 * TILE];  // rows: 0-1 score, 2-5 bbox, 6-9 mask(2:6), 10-15 label
  __shared__ float s_f[3];
  __shared__ unsigned s_u[2];

  const int blk = blockIdx.x;
  const int b   = blk / chunks;
  const int ck  = blk % chunks;
  const int hw0 = ck * TILE;
  const int i   = threadIdx.x;

  if (i == 0) { s_f[0] = s_f[1] = s_f[2] = 0.0f; s_u[0] = s_u[1] = 0u; }

  const float* ps_base = ps + ((size_t)b * 2) * HW + hw0;
  const float* pb_base = pb + ((size_t)b * 4) * HW + hw0;
  const float* gm_base = gm + ((size_t)b * 6 + 2) * HW + hw0;
  const float* gl_base = gl + ((size_t)b * 6) * HW + hw0;

#ifdef HAVE_TDM
  if (i < 32) {  // wave 0 only issues the DMA (uniform branch; other waves skip)
    unsigned lds0 = (unsigned)(uintptr_t)&tile[0];
    tdm_load_2d_f32(lds0,                  ps_base, TILE, 2, (unsigned long long)HW);
    tdm_load_2d_f32(lds0 + 2  * TILE * 4,  pb_base, TILE, 4, (unsigned long long)HW);
    tdm_load_2d_f32(lds0 + 6  * TILE * 4,  gm_base, TILE, 4, (unsigned long long)HW);
    tdm_load_2d_f32(lds0 + 10 * TILE * 4,  gl_base, TILE, 6, (unsigned long long)HW);
    wait_tensorcnt0();
  }
#else
#pragma unroll
  for (int r = 0; r < 16; ++r) {
    const float* src = r < 2  ? ps_base + (size_t)r * HW
                     : r < 6  ? pb_base + (size_t)(r - 2) * HW
                     : r < 10 ? gm_base + (size_t)(r - 6) * HW
                              : gl_base + (size_t)(r - 10) * HW;
    tile[r * TILE + i] = src[i];
  }
#endif
  __syncthreads();

  const float s0 = tile[i];
  const float s1 = tile[TILE + i];
  const float mx = fmaxf(s0, s1);
  const float e0 = expf(s0 - mx);
  const float e1 = expf(s1 - mx);
  const float inv = 1.0f / (e0 + e1);
  const float p0 = e0 * inv;
  const float p1 = e1 * inv;  // p_neg

  const float gl0 = tile[10 * TILE + i];
  const float gl1 = tile[11 * TILE + i];
  const bool pos = gl0 > 0.5f;
  const bool neg = gl1 > 0.5f;

  float spce = pos ? (-gl0 * logf(p0)) : 0.0f;

  packed[(size_t)b * HW + hw0 + i] =
      (__float_as_uint(p1) & 0x7FFFFFFFu) | (neg ? 0x80000000u : 0u);

  float sm = 0.0f, ssq = 0.0f;
#pragma unroll
  for (int c = 0; c < 4; ++c) {
    const float mb = tile[(6 + c) * TILE + i];
    const float d  = (tile[(2 + c) * TILE + i] - tile[(12 + c) * TILE + i]) * mb;
    sm  += mb;
    ssq += d * d;
  }

  unsigned pc = pos ? 1u : 0u;
  unsigned nc = neg ? 1u : 0u;
#pragma unroll
  for (int off = 16; off > 0; off >>= 1) {
    spce += __shfl_down(spce, off, 32);
    sm   += __shfl_down(sm,   off, 32);
    ssq  += __shfl_down(ssq,  off, 32);
    pc   += __shfl_down(pc,   off, 32);
    nc   += __shfl_down(nc,   off, 32);
  }
  if ((threadIdx.x & 31) == 0) {
    atomicAdd(&s_f[0], spce); atomicAdd(&s_f[1], sm); atomicAdd(&s_f[2], ssq);
    atomicAdd(&s_u[0], pc);   atomicAdd(&s_u[1], nc);
  }
  __syncthreads();
  if (threadIdx.x == 0) {
    atomicAdd(&st->sum_pos_ce, s_f[0]);
    atomicAdd(&st->sum_mask,   s_f[1]);
    atomicAdd(&st->sum_sq,     s_f[2]);
    atomicAdd(&st->pos_num,    s_u[0]);
    atomicAdd(&st->neg_num,    s_u[1]);
  }
}

// Decide rank & zero shortcut for the k-th smallest selection.
__global__ void prepare_kernel(BranchStats* st, int N, int kNoPos) {
  if (threadIdx.x == 0 && blockIdx.x == 0) {
    const unsigned pos = st->pos_num, neg = st->neg_num;
    const bool has_pos = pos > 0u;
    unsigned long long k;
    if (has_pos) {
      unsigned long long kp = 5ull * (unsigned long long)pos;
      k = kp < (unsigned long long)neg ? kp : (unsigned long long)neg;
    } else {
      k = (unsigned long long)kNoPos;
    }
    // threshold = sorted[clip(k-1, 0, N-1)] -> rank r = clip(k, 1, N) (1-based)
    unsigned long long r = k;
    if (r < 1ull) r = 1ull;
    if (r > (unsigned long long)N) r = (unsigned long long)N;
    const unsigned long long zeros = has_pos ? (unsigned long long)(N - (long long)neg) : 0ull;
    if (r <= zeros) {
      st->thr_bits = 0u;   // k-th smallest is one of the zero candidates
      st->done = 1u;
    } else {
      st->rr = (unsigned)(r - zeros);  // rank among strictly-positive candidates
      st->prefix = 0u;
      st->done = 0u;
    }
  }
}

// Radix-select histogram pass over 31-bit non-negative float candidates.
// pass 0: bits[30:20] (2048 bins), pass 1: bits[19:10], pass 2: bits[9:0].
__global__ __launch_bounds__(256) void hist_kernel(
    const unsigned* __restrict__ packed, int N,
    const BranchStats* __restrict__ st, unsigned* __restrict__ hist, int pass) {
  __shared__ unsigned lh[2048];
  if (st->done) return;  // uniform across grid
  const bool has_pos = st->pos_num > 0u;
  const unsigned prefix = st->prefix;
  const int nbins = (pass == 0) ? 2048 : 1024;
  for (int t = threadIdx.x; t < nbins; t += blockDim.x) lh[t] = 0u;
  __syncthreads();
  const int stride = gridDim.x * blockDim.x;
  for (int idx = blockIdx.x * blockDim.x + threadIdx.x; idx < N; idx += stride) {
    __builtin_prefetch(packed + idx + stride, 0, 0);
    const unsigned pk = packed[idx];
    const unsigned pv = pk & 0x7FFFFFFFu;
    const bool negf = (pk >> 31) != 0u;
    const unsigned c = has_pos ? (negf ? pv : 0u) : pv;
    if (c == 0u) continue;  // zeros handled analytically
    if (pass == 0) {
      atomicAdd(&lh[c >> 20], 1u);
    } else if (pass == 1) {
      if ((c >> 20) == prefix) atomicAdd(&lh[(c >> 10) & 1023u], 1u);
    } else {
      if ((c >> 10) == prefix) atomicAdd(&lh[c & 1023u], 1u);
    }
  }
  __syncthreads();
  for (int t = threadIdx.x; t < nbins; t += blockDim.x) {
    const unsigned v = lh[t];
    if (v) atomicAdd(&hist[t], v);
  }
}

// Pick the bin containing the remaining rank, update prefix, and rezero hist.
__global__ __launch_bounds__(256) void scan_kernel(BranchStats* st, unsigned* hist, int pass) {
  if (threadIdx.x == 0 && blockIdx.x == 0) {
    if (!st->done) {
      const int nbins = (pass == 0) ? 2048 : 1024;
      unsigned long long cum = 0;
      const unsigned rr = st->rr;
      int sel = nbins - 1;
      for (int b = 0; b < nbins; ++b) {
        const unsigned v = hist[b];
        if (cum + v >= (unsigned long long)rr) { sel = b; st->rr = (unsigned)(rr - cum); break; }
        cum += v;
      }
      if (pass == 0)      st->prefix = (unsigned)sel;
      else if (pass == 1) st->prefix = (st->prefix << 10) | (unsigned)sel;
      else { st->thr_bits = (st->prefix << 10) | (unsigned)sel; st->done = 1u; }
    }
  }
  __syncthreads();
  for (int t = threadIdx.x; t < 2048; t += blockDim.x) hist[t] = 0u;
}

// Final pass: count neg_prob <= thr (ties included) and sum -gl1*log(p1).
__global__ __launch_bounds__(256) void finalneg_kernel(
    const unsigned* __restrict__ packed, const float* __restrict__ gl,
    BranchStats* __restrict__ st, int HW) {
  __shared__ float s_ce;
  __shared__ unsigned s_cnt;
  if (threadIdx.x == 0) { s_ce = 0.0f; s_cnt = 0u; }
  __syncthreads();
  const int b = blockIdx.y;
  const int hw = blockIdx.x * blockDim.x + threadIdx.x;
  const bool has_pos = st->pos_num > 0u;
  const float thr = __uint_as_float(st->thr_bits);
  float ce = 0.0f;
  unsigned cnt = 0u;
  if (hw < HW) {
    const unsigned pk = packed[(size_t)b * HW + hw];
    const float p1 = __uint_as_float(pk & 0x7FFFFFFFu);
    const bool negf = (pk >> 31) != 0u;
    const float npv = has_pos ? (negf ? p1 : 0.0f) : p1;
    if (npv <= thr) {
      const float g1 = gl[((size_t)b * 6 + 1) * HW + hw];
      ce = -g1 * logf(p1);
      cnt = 1u;
    }
  }
#pragma unroll
  for (int off = 16; off > 0; off >>= 1) {
    ce  += __shfl_down(ce,  off, 32);
    cnt += __shfl_down(cnt, off, 32);
  }
  if ((threadIdx.x & 31) == 0) { atomicAdd(&s_ce, ce); atomicAdd(&s_cnt, cnt); }
  __syncthreads();
  if (threadIdx.x == 0) {
    atomicAdd(&st->sum_neg_ce, s_ce);
    atomicAdd(&st->cnt_neg, s_cnt);
  }
}

__global__ void finalize_kernel(const BranchStats* __restrict__ st, float* __restrict__ out) {
  if (threadIdx.x == 0 && blockIdx.x == 0) {
    float r[4];
    for (int i = 0; i < 2; ++i) {
      const BranchStats s = st[i];
      const unsigned cnt = s.pos_num + s.cnt_neg;
      const float denom = (float)(cnt > 0u ? cnt : 1u);
      const float ls = (s.sum_pos_ce + s.sum_neg_ce) / denom;
      const float lb = (s.sum_mask > 0.0f) ? (s.sum_sq / fmaxf(s.sum_mask, 1e-8f)) : 0.0f;
      r[2 * i] = ls;
      r[2 * i + 1] = lb;
    }
    out[0] = r[0] + r[1] + r[2] + r[3];
    out[1] = r[0]; out[2] = r[1]; out[3] = r[2]; out[4] = r[3];
  }
}

extern "C" void kernel_launch(void* const* d_in, const int* in_sizes, int n_in,
                              void* d_out, int out_size, void* d_ws, size_t ws_size,
                              hipStream_t stream) {
  (void)in_sizes; (void)n_in; (void)out_size; (void)ws_size;
  const float* ps0 = (const float*)d_in[0];
  const float* pb0 = (const float*)d_in[1];
  const float* gm0 = (const float*)d_in[2];
  const float* gl0 = (const float*)d_in[3];
  const float* ps1 = (const float*)d_in[4];
  const float* pb1 = (const float*)d_in[5];
  const float* gm1 = (const float*)d_in[6];
  const float* gl1 = (const float*)d_in[7];

  const int B = 32;
  const int HW0 = 320 * 320, HW1 = 160 * 160;
  const int N0 = B * HW0, N1 = B * HW1;

  unsigned* packed0 = (unsigned*)d_ws;                       // N0 u32
  unsigned* packed1 = packed0 + N0;                          // N1 u32
  BranchStats* st   = (BranchStats*)(packed1 + N1);          // 2 structs
  unsigned* hist    = (unsigned*)(st + 2);                   // 2048 u32

  init_kernel<<<8, 256, 0, stream>>>((unsigned*)st, 2 * STATS_U32 + 2048);

  pass1_kernel<<<B * (HW0 / TILE), TILE, 0, stream>>>(ps0, pb0, gm0, gl0, packed0, st + 0, HW0, HW0 / TILE);
  pass1_kernel<<<B * (HW1 / TILE), TILE, 0, stream>>>(ps1, pb1, gm1, gl1, packed1, st + 1, HW1, HW1 / TILE);

  prepare_kernel<<<1, 1, 0, stream>>>(st + 0, N0, N0 / 10);  // int(N*0.1)
  prepare_kernel<<<1, 1, 0, stream>>>(st + 1, N1, N1 / 10);

  for (int pass = 0; pass < 3; ++pass) {
    hist_kernel<<<2048, 256, 0, stream>>>(packed0, N0, st + 0, hist, pass);
    scan_kernel<<<1, 256, 0, stream>>>(st + 0, hist, pass);
  }
  for (int pass = 0; pass < 3; ++pass) {
    hist_kernel<<<1024, 256, 0, stream>>>(packed1, N1, st + 1, hist, pass);
    scan_kernel<<<1, 256, 0, stream>>>(st + 1, hist, pass);
  }

  finalneg_kernel<<<dim3(HW0 / 256, B), 256, 0, stream>>>(packed0, gl0, st + 0, HW0);
  finalneg_kernel<<<dim3(HW1 / 256, B), 256, 0, stream>>>(packed1, gl1, st + 1, HW1);

  finalize_kernel<<<1, 1, 0, stream>>>(st, (float*)d_out);
}